// GraphConvLayer_39797166965436
// MI455X (gfx1250) — compile-verified
//
#include <hip/hip_runtime.h>

#define N_NODES 50000
#define N_EDGES 800000
#define DIM 128

typedef __attribute__((ext_vector_type(2))) float v2f;
typedef __attribute__((ext_vector_type(8))) float v8f;

// Native fp32 atomic add at device scope via inline asm so the compiler cannot
// legalize it into a CAS loop. No-return form -> tracked with STOREcnt;
// s_endpgm's implicit wait-idle covers completion at kernel end.
__device__ __forceinline__ void atomic_add_f32_dev(float* addr, float val) {
  asm volatile("global_atomic_add_f32 %0, %1, off scope:SCOPE_DEV"
               :: "v"((unsigned long long)addr), "v"(val)
               : "memory");
}

// Fast reciprocal: v_rcp_f32 + one Newton-Raphson step (~0.5 ulp), avoiding
// the IEEE div_scale/div_fmas/div_fixup chain in the GEMM prologue.
__device__ __forceinline__ float fast_rcp(float d) {
  float r = __builtin_amdgcn_rcpf(d);
  return r * (2.0f - d * r);
}

// ---------------------------------------------------------------- zero agg+deg
__global__ void zero_ws_kernel(float* __restrict__ p, int total) {
  int i = blockIdx.x * blockDim.x + threadIdx.x;
  const int stride = gridDim.x * blockDim.x;
  for (; i < total; i += stride) p[i] = 0.0f;
}

// ---------------------------------------------------------------- scatter-add
// One wave32 per edge. Lane L handles features [4L, 4L+4): float4 gather from
// x[src], 4 native fp32 atomics into agg[dst] (L2-resident, 25.6MB < 192MB L2).
__global__ void __launch_bounds__(256)
scatter_kernel(const float* __restrict__ x,
               const int* __restrict__ edge_index,
               float* __restrict__ agg,
               float* __restrict__ deg) {
  const int lane = threadIdx.x & 31;
  const int edge = (blockIdx.x * blockDim.x + threadIdx.x) >> 5;
  if (edge >= N_EDGES) return;
  const int src = edge_index[edge];            // row 0 of [2,E]
  const int dst = edge_index[N_EDGES + edge];  // row 1 of [2,E]
  const float4 v = *(const float4*)(x + (size_t)src * DIM + lane * 4);
  float* a = agg + (size_t)dst * DIM + lane * 4;
  atomic_add_f32_dev(a + 0, v.x);
  atomic_add_f32_dev(a + 1, v.y);
  atomic_add_f32_dev(a + 2, v.z);
  atomic_add_f32_dev(a + 3, v.w);
  if (lane == 0) atomic_add_f32_dev(deg + dst, 1.0f);
}

// ---------------------------------------------------------------- fused GEMM
// out = x @ w_self^T + b_self + (agg/deg) @ w_neigh^T + b_neigh
// Block = 256 threads = 8 waves; block covers a 16-row tile, wave w covers
// output columns [16w, 16w+16). Both GEMMs chained into one f32 accumulator
// with V_WMMA_F32_16X16X4_F32 (exact fp32).
//
// Fragment layout (CDNA5 ISA, 32-bit 16x4 A): lane L -> M = L%16,
// Kbase = 2*(L/16); A = {A[M,k+Kb], A[M,k+Kb+1]}. B = W^T means
// B[k][n] = w[n][k], so the B fragment uses the identical addressing on w.
// C/D: VGPR v -> row M = v + 8*(L/16), col N = L%16.
__global__ void __launch_bounds__(256)
fused_gemm_kernel(const float* __restrict__ x,
                  const float* __restrict__ agg,
                  const float* __restrict__ deg,
                  const float* __restrict__ w_self,
                  const float* __restrict__ b_self,
                  const float* __restrict__ w_neigh,
                  const float* __restrict__ b_neigh,
                  float* __restrict__ out) {
  const int lane  = threadIdx.x & 31;
  const int wave  = threadIdx.x >> 5;
  const int m     = lane & 15;
  const int khalf = lane >> 4;  // 0 or 1
  const int row0  = blockIdx.x * 16;
  const int arow  = row0 + m;        // A-fragment row this lane serves
  const int ncol  = wave * 16 + m;   // output column this lane serves

  // Fold mean normalization into the A_neigh load.
  const float invd = fast_rcp(fmaxf(deg[arow], 1.0f));

  const float* xrow  = x      + (size_t)arow * DIM + khalf * 2;
  const float* grow  = agg    + (size_t)arow * DIM + khalf * 2;
  const float* wsrow = w_self + (size_t)ncol * DIM + khalf * 2;
  const float* wnrow = w_neigh+ (size_t)ncol * DIM + khalf * 2;

  v8f c = {};
#pragma unroll
  for (int k = 0; k < DIM; k += 4) {
    v2f a_s = *(const v2f*)(xrow + k);
    v2f b_s = *(const v2f*)(wsrow + k);
    v2f g   = *(const v2f*)(grow + k);
    v2f b_n = *(const v2f*)(wnrow + k);
    v2f a_n;
    a_n.x = g.x * invd;
    a_n.y = g.y * invd;
    // 8 args: (neg_a, A, neg_b, B, c_mod, C, reuse_a, reuse_b)
    c = __builtin_amdgcn_wmma_f32_16x16x4_f32(false, a_s, false, b_s,
                                              (short)0, c, false, false);
    c = __builtin_amdgcn_wmma_f32_16x16x4_f32(false, a_n, false, b_n,
                                              (short)0, c, false, false);
  }

  const float bias = b_self[ncol] + b_neigh[ncol];
  float* orow = out + (size_t)(row0 + khalf * 8) * DIM + ncol;
#pragma unroll
  for (int v = 0; v < 8; ++v) {
    orow[(size_t)v * DIM] = c[v] + bias;
  }
}

// ---------------------------------------------------------------- launch
extern "C" void kernel_launch(void* const* d_in, const int* in_sizes, int n_in,
                              void* d_out, int out_size, void* d_ws, size_t ws_size,
                              hipStream_t stream) {
  const float* x       = (const float*)d_in[0];
  const int*   ei      = (const int*)  d_in[1];
  const float* w_self  = (const float*)d_in[2];
  const float* b_self  = (const float*)d_in[3];
  const float* w_neigh = (const float*)d_in[4];
  const float* b_neigh = (const float*)d_in[5];
  float* out = (float*)d_out;

  float* agg = (float*)d_ws;                      // N*DIM floats
  float* deg = agg + (size_t)N_NODES * DIM;       // N floats

  const int ztotal = N_NODES * DIM + N_NODES;
  zero_ws_kernel<<<512, 256, 0, stream>>>(agg, ztotal);

  scatter_kernel<<<(N_EDGES * 32) / 256, 256, 0, stream>>>(x, ei, agg, deg);

  fused_gemm_kernel<<<N_NODES / 16, 256, 0, stream>>>(
      x, agg, deg, w_self, b_self, w_neigh, b_neigh, out);
}